// gmm_layer_57174604645138
// MI455X (gfx1250) — compile-verified
//
#include <hip/hip_runtime.h>
#include <math.h>

// MI455X / gfx1250, wave32. D = A(16x4,f32) x B(4x16,f32) + C(16x16,f32)
typedef __attribute__((ext_vector_type(2))) float v2f;
typedef __attribute__((ext_vector_type(8))) float v8f;

#define DIMD 256
#define NCLS 256
#define NBATCH 4096

// Wt layout (e-chunk interleaved for single-ds_load_b64 B operands):
//   Wt[((k*64 + ec)*256 + i)*4 + el]  ==  W[i][e]  with e = ec*4+el,
//   W[i][e] = L_inv[i][e] * rsqrt(D_i)

// ---------------------------------------------------------------------------
// Kernel 1: per-class unit-lower-triangular inversion + row scaling.
// Solves L * X = I by forward substitution; thread j owns column c = c0+j.
// X column-per-thread in LDS, pitch 65 (bank (m+j)%64: conflict-free).
// grid (4, C), block 64 (2 waves).
// ---------------------------------------------------------------------------
__global__ __launch_bounds__(64, 1)
void tri_invert_kernel(const float* __restrict__ Lparam,
                       float* __restrict__ Wt,
                       float* __restrict__ coeff) {
  __shared__ float X[DIMD * 65];   // 66.5 KB
  __shared__ float rowbuf[DIMD];
  __shared__ float drs[DIMD];
  const int j  = threadIdx.x;      // 0..63
  const int g  = blockIdx.x;       // column group
  const int k  = blockIdx.y;       // class
  const int c0 = g * 64;
  const int c  = c0 + j;           // my column (= e index)
  const float* Lk = Lparam + (size_t)k * DIMD * DIMD;

  // stage rsqrt(D_m), D_m = diag(Lparam)^2 + 1e-4
  for (int q = 0; q < 4; ++q) {
    const int m = j + q * 64;
    const float dg = Lk[(size_t)m * DIMD + m];
    drs[m] = rsqrtf(dg * dg + 1e-4f);
  }
  __syncthreads();
  if (g == 0 && j == 0) {          // det_scale = prod(rsqrt(D)) = rsqrt(prod D)
    float p = 1.0f;
    for (int m = 0; m < DIMD; ++m) p *= drs[m];
    coeff[k] = p;
  }

  // forward substitution (unit diagonal): rows < c0 never read by this group
  for (int i = c0; i < DIMD; ++i) {
    __syncthreads();                               // protect rowbuf
    for (int q = 0; q < 4; ++q)
      rowbuf[j + q * 64] = Lk[(size_t)i * DIMD + j + q * 64];
    __syncthreads();
    float s = 0.0f;
    for (int m = c0; m < i; ++m)                   // rowbuf[m]: broadcast read
      s += rowbuf[m] * X[m * 65 + j];              // X: conflict-free
    X[i * 65 + j] = ((i == c) ? 1.0f : 0.0f) - s;  // rows in [c0,c) become 0
  }
  __syncthreads();

  // write interleaved layout: Wt[((k*64 + (c>>2))*256 + m)*4 + (c&3)]
  float* dstk = Wt + (((size_t)k * 64 + (c >> 2)) * DIMD + 0) * 4 + (c & 3);
  for (int m = 0; m < DIMD; ++m) {
    const float xv = (m >= c) ? X[m * 65 + j] : 0.0f;  // strict upper = 0
    dstk[m * 4] = xv * drs[m];
  }
}

// ---------------------------------------------------------------------------
// Kernel 2: expo[b][k] = -0.5 * || W_k (x_b - c_k) ||^2  via f32 WMMA.
// grid (B/64, C), block 128 (4 waves); wave w owns batch rows rb..rb+15.
// Per e-chunk of 4: stage the contiguous 4KB slab Wt[k][ec][:][:] to LDS,
// build the A operand (diff) once, issue 16 v_wmma_f32_16x16x4_f32 into 16
// register-resident accumulator tiles (128 VGPRs; launch_bounds prevents
// the spills seen with the default occupancy heuristic).
// A layout: lanes0-15 {K=0,1} row M=lane; lanes16-31 {K=2,3} row M=lane-16.
// B layout: lanes0-15 {K=0,1} col N=lane; lanes16-31 {K=2,3} col N=lane-16.
//   -> B pair = one ds_load_b64 at wbuf[i*4 + half*2] (banks disjoint per half)
// D layout: VGPR r -> M=r (lanes0-15) / M=r+8 (lanes16-31), N=lane%16.
// ---------------------------------------------------------------------------
__global__ __launch_bounds__(128, 1)
void mahal_kernel(const float* __restrict__ x,
                  const float* __restrict__ centers,
                  const float* __restrict__ Wt,
                  float* __restrict__ expo) {
  __shared__ float wbuf[DIMD * 4];   // 4 KB: [i][el]
  __shared__ float cbuf[DIMD];       // centers[k]
  const int t    = threadIdx.x;      // 0..127
  const int k    = blockIdx.y;
  const int wave = t >> 5;
  const int lane = t & 31;
  const int row  = lane & 15;
  const int half = lane >> 4;
  const int rb   = blockIdx.x * 64 + wave * 16;

  cbuf[t]       = centers[(size_t)k * DIMD + t];
  cbuf[t + 128] = centers[(size_t)k * DIMD + t + 128];

  v8f acc[16] = {};                          // 16 i-tiles resident
  const float* Wk   = Wt + (size_t)k * 64 * DIMD * 4;
  const float* xrow = x + (size_t)(rb + row) * DIMD;

  for (int ec = 0; ec < 64; ++ec) {
    const int e0 = ec * 4;
    __syncthreads();                         // slab consumers done
    const float4* src  = (const float4*)(Wk + (size_t)ec * DIMD * 4);
    float4*       dst4 = (float4*)wbuf;
    dst4[t]       = src[t];                  // 4KB contiguous, coalesced
    dst4[t + 128] = src[t + 128];
    __syncthreads();

    const int eA = e0 + half * 2;            // A operand: diff on the fly
    const v2f xa = *(const v2f*)(xrow + eA);
    const v2f ca = *(const v2f*)(cbuf + eA);
    const v2f a  = xa - ca;

    const float* wb = wbuf + half * 2;       // K-pair base for this half
#pragma unroll
    for (int ii = 0; ii < 16; ++ii) {
      const int i = ii * 16 + row;           // N index
      const v2f b = *(const v2f*)(wb + i * 4);   // single ds_load_b64
      acc[ii] = __builtin_amdgcn_wmma_f32_16x16x4_f32(
          false, a, false, b, (short)0, acc[ii], false, false);
    }
  }

  // dist[M] = sum over all 256 i of Y[M,i]^2: square locally, butterfly over
  // the 16 lanes of each half (xor masks < 16 stay within the half).
  float ds[8];
#pragma unroll
  for (int r = 0; r < 8; ++r) {
    float s = 0.0f;
#pragma unroll
    for (int ii = 0; ii < 16; ++ii) { const float y = acc[ii][r]; s += y * y; }
    ds[r] = s;
  }
#pragma unroll
  for (int r = 0; r < 8; ++r)
#pragma unroll
    for (int mk = 1; mk < 16; mk <<= 1)
      ds[r] += __shfl_xor(ds[r], mk, 32);

  if ((lane & 15) == 0) {                    // lane 0 -> M=r, lane 16 -> M=r+8
    const int M0 = rb + half * 8;
#pragma unroll
    for (int r = 0; r < 8; ++r)
      expo[(size_t)(M0 + r) * NCLS + k] = -0.5f * ds[r];
  }
}

// ---------------------------------------------------------------------------
// Kernel 3: resp[b][k] = coeff[k]*exp(expo - max) / sum_k(...). grid B, block C.
// ---------------------------------------------------------------------------
__global__ __launch_bounds__(256)
void softmax_kernel(const float* __restrict__ expo,
                    const float* __restrict__ coeff,
                    float* __restrict__ out) {
  __shared__ float red[NCLS];
  const int b = blockIdx.x;
  const int t = threadIdx.x;
  const float e = expo[(size_t)b * NCLS + t];
  red[t] = e;
  __syncthreads();
  for (int s = NCLS / 2; s > 0; s >>= 1) {
    if (t < s) red[t] = fmaxf(red[t], red[t + s]);
    __syncthreads();
  }
  const float mx = red[0];
  __syncthreads();
  const float numer = coeff[t] * expf(e - mx);
  red[t] = numer;
  __syncthreads();
  for (int s = NCLS / 2; s > 0; s >>= 1) {
    if (t < s) red[t] += red[t + s];
    __syncthreads();
  }
  out[(size_t)b * NCLS + t] = numer / red[0];
}

// ---------------------------------------------------------------------------
extern "C" void kernel_launch(void* const* d_in, const int* in_sizes, int n_in,
                              void* d_out, int out_size, void* d_ws, size_t ws_size,
                              hipStream_t stream) {
  (void)in_sizes; (void)n_in; (void)out_size; (void)ws_size;
  const float* x       = (const float*)d_in[0];
  const float* centers = (const float*)d_in[1];
  const float* Lparam  = (const float*)d_in[2];
  float* out = (float*)d_out;

  // workspace: Wt [C][64][256][4] f32 (64 MB, resident in 192MB L2 for the
  // GEMM phase), coeff [C], expo [B][C]  -> ~71.3 MB total
  float* Wt    = (float*)d_ws;
  float* coeff = Wt + (size_t)NCLS * DIMD * DIMD;
  float* expo  = coeff + NCLS;

  tri_invert_kernel<<<dim3(4, NCLS), 64, 0, stream>>>(Lparam, Wt, coeff);
  mahal_kernel<<<dim3(NBATCH / 64, NCLS), 128, 0, stream>>>(x, centers, Wt, expo);
  softmax_kernel<<<NBATCH, NCLS, 0, stream>>>(expo, coeff, out);
}